// GIN_13889924235785
// MI455X (gfx1250) — compile-verified
//
#include <hip/hip_runtime.h>
#include <hip/hip_bf16.h>
#include <stdint.h>

typedef __attribute__((ext_vector_type(2))) float v2f;
typedef __attribute__((ext_vector_type(8))) float v8f;

#define GIN_N_NODES   50000
#define GIN_N_EDGES   800000
#define GIN_HID       128
#define GIN_N_CLASSES 10
#define GIN_N_GRAPHS  64
#define GIN_ZPAD      132   // A-tile LDS row pitch (floats): 16B-aligned, stride 4 banks
#define GIN_TILES_M   5     // M-tiles (of 16 rows) per block; 50000 = 625 * 80
#define GIN_ROWS_BLK  (GIN_TILES_M * 16)

// ---------------------------------------------------------------------------
// zero fill (float4 granularity; all our buffers are multiples of 4 floats)
// ---------------------------------------------------------------------------
__global__ void gin_zero_f32(float* __restrict__ p, int n4) {
  int i = blockIdx.x * blockDim.x + threadIdx.x;
  int stride = gridDim.x * blockDim.x;
  float4 z = make_float4(0.f, 0.f, 0.f, 0.f);
  for (; i < n4; i += stride) ((float4*)p)[i] = z;
}

__device__ __forceinline__ void gin_atomic_add(float* p, float v) {
  __hip_atomic_fetch_add(p, v, __ATOMIC_RELAXED, __HIP_MEMORY_SCOPE_AGENT);
}

// ---------------------------------------------------------------------------
// agg[dst] += H[src]   (one wave per edge, float4 per lane = 128 channels)
// ---------------------------------------------------------------------------
__global__ void gin_scatter_edges(const float* __restrict__ H,
                                  const int* __restrict__ src,
                                  const int* __restrict__ dst,
                                  float* __restrict__ AGG) {
  int gw   = (blockIdx.x * blockDim.x + threadIdx.x) >> 5;
  int lane = threadIdx.x & 31;
  int nw   = (gridDim.x * blockDim.x) >> 5;
  for (int e = gw; e < GIN_N_EDGES; e += nw) {
    int s = src[e];
    int d = dst[e];
    float4 v = *(const float4*)(H + (size_t)s * GIN_HID + lane * 4);
    float* p = AGG + (size_t)d * GIN_HID + lane * 4;
    gin_atomic_add(p + 0, v.x);
    gin_atomic_add(p + 1, v.y);
    gin_atomic_add(p + 2, v.z);
    gin_atomic_add(p + 3, v.w);
  }
}

// ---------------------------------------------------------------------------
// OUT = relu( (X [+ AGG]) @ W + bias ),  X:[M,128], W:[128,128] (KxN)
//
// block = 256 threads (8 wave32), 80 rows (5 M-tiles) per block, wave w owns
// cols [16w, 16w+16) across all 5 M-tiles.
//   * W (64 KB) staged via gfx1250 async global->LDS engine
//     (global_load_async_to_lds_b128 + s_wait_asynccnt fence); amortized
//     over 80 output rows.
//   * A-tile (80x128, GIN "x + agg" fused once at stage time) staged via the
//     DS path, overlapping the async W copy.
//   * Inner loop per k-step: one B-fragment gather feeding 5 WMMAs on 5
//     INDEPENDENT accumulator chains (hides v_wmma RAW latency).
//
// fp32 WMMA operand layout (CDNA5 ISA 7.12.2):
//   A (16x4):  lane L holds row M=L&15, K = {2*(L>>4), 2*(L>>4)+1}   -> v2f
//   B (4x16):  lane L holds col N=L&15, K = {2*(L>>4), 2*(L>>4)+1}   -> v2f
//   C/D:       vgpr r, lane L -> row r + 8*(L>>4), col L&15          -> v8f
// ---------------------------------------------------------------------------
template <bool FUSE_ADD>
__global__ void gin_gemm_relu(const float* __restrict__ X,
                              const float* __restrict__ AGG,
                              const float* __restrict__ W,
                              const float* __restrict__ bias,
                              float* __restrict__ OUT) {
  __shared__ float sZ[GIN_ROWS_BLK * GIN_ZPAD];  // 42.2 KB A-tile (low offsets)
  __shared__ float sW[GIN_HID * GIN_HID];        // 64 KB, row-major [k][n]

  const int tid  = threadIdx.x;
  const int row0 = blockIdx.x * GIN_ROWS_BLK;

  // ---- async-stage W: 16 B per lane per issue, 16 issues per thread ----
#pragma unroll
  for (int i = tid * 4; i < GIN_HID * GIN_HID; i += 256 * 4) {
    uint32_t lds_off = (uint32_t)(uintptr_t)(sW + i);  // flat[31:0] == LDS offset
    const float* gsrc = W + i;
    asm volatile("global_load_async_to_lds_b128 %0, %1, off"
                 :
                 : "v"(lds_off), "v"(gsrc)
                 : "memory");
  }

  // ---- stage A-tile through DS path (overlaps the async W copy) ----
  for (int i = tid * 4; i < GIN_ROWS_BLK * GIN_HID; i += 256 * 4) {
    const int r = i >> 7;
    const int c = i & (GIN_HID - 1);
    float4 xv = *(const float4*)(X + (size_t)(row0 + r) * GIN_HID + c);
    if (FUSE_ADD) {
      float4 av = *(const float4*)(AGG + (size_t)(row0 + r) * GIN_HID + c);
      xv.x += av.x; xv.y += av.y; xv.z += av.z; xv.w += av.w;
    }
    *(float4*)(sZ + r * GIN_ZPAD + c) = xv;
  }

  asm volatile("s_wait_asynccnt 0x0" ::: "memory");
  __syncthreads();

  const int wave  = tid >> 5;
  const int lane  = tid & 31;
  const int col0  = wave * 16;
  const int lrow  = lane & 15;        // M (A-frag) / N (B,C-frag) within tile
  const int khalf = (lane >> 4) * 2;  // K pair selector: 0 or 2

  v8f c[GIN_TILES_M];
  const float bv = bias[col0 + lrow];
#pragma unroll
  for (int t = 0; t < GIN_TILES_M; ++t)
#pragma unroll
    for (int r = 0; r < 8; ++r) c[t][r] = bv;

  const float* zrow = sZ + lrow * GIN_ZPAD;
  const float* wcol = sW + col0 + lrow;

#pragma unroll
  for (int k = 0; k < GIN_HID; k += 4) {
    const int kk = k + khalf;
    v2f b;  // one B fragment feeds all 5 M-tiles
    b[0] = wcol[(kk + 0) * GIN_HID];
    b[1] = wcol[(kk + 1) * GIN_HID];
#pragma unroll
    for (int t = 0; t < GIN_TILES_M; ++t) {
      v2f a = *(const v2f*)(zrow + t * (16 * GIN_ZPAD) + kk);  // ds_load_b64
      c[t] = __builtin_amdgcn_wmma_f32_16x16x4_f32(false, a, false, b,
                                                   (short)0, c[t], false, false);
    }
  }

#pragma unroll
  for (int t = 0; t < GIN_TILES_M; ++t) {
    const int rbase = row0 + t * 16 + (lane >> 4) * 8;
#pragma unroll
    for (int r = 0; r < 8; ++r) {
      float v = c[t][r];
      v = v > 0.f ? v : 0.f;
      OUT[(size_t)(rbase + r) * GIN_HID + col0 + lrow] = v;
    }
  }
}

// ---------------------------------------------------------------------------
// G[batch[n]] += H[n]   (one wave per node)
// ---------------------------------------------------------------------------
__global__ void gin_pool(const float* __restrict__ H,
                         const int* __restrict__ batch,
                         float* __restrict__ G) {
  int gw   = (blockIdx.x * blockDim.x + threadIdx.x) >> 5;
  int lane = threadIdx.x & 31;
  int nw   = (gridDim.x * blockDim.x) >> 5;
  for (int n = gw; n < GIN_N_NODES; n += nw) {
    int b = batch[n];
    float4 v = *(const float4*)(H + (size_t)n * GIN_HID + lane * 4);
    float* p = G + (size_t)b * GIN_HID + lane * 4;
    gin_atomic_add(p + 0, v.x);
    gin_atomic_add(p + 1, v.y);
    gin_atomic_add(p + 2, v.z);
    gin_atomic_add(p + 3, v.w);
  }
}

// ---------------------------------------------------------------------------
// out[64,10] = G[64,128] @ WC[128,10] + bc   (tiny; one thread per output)
// ---------------------------------------------------------------------------
__global__ void gin_classifier(const float* __restrict__ G,
                               const float* __restrict__ WC,
                               const float* __restrict__ bc,
                               float* __restrict__ out) {
  int t = blockIdx.x * blockDim.x + threadIdx.x;
  if (t < GIN_N_GRAPHS * GIN_N_CLASSES) {
    int g = t / GIN_N_CLASSES;
    int cc = t % GIN_N_CLASSES;
    float acc = bc[cc];
    const float* grow = G + (size_t)g * GIN_HID;
#pragma unroll 8
    for (int k = 0; k < GIN_HID; ++k) acc += grow[k] * WC[k * GIN_N_CLASSES + cc];
    out[t] = acc;
  }
}

// ---------------------------------------------------------------------------
extern "C" void kernel_launch(void* const* d_in, const int* in_sizes, int n_in,
                              void* d_out, int out_size, void* d_ws,
                              size_t ws_size, hipStream_t stream) {
  const float* x    = (const float*)d_in[0];
  const int* eidx   = (const int*)d_in[1];
  const int* batch  = (const int*)d_in[2];
  const float* w1a  = (const float*)d_in[3];
  const float* b1a  = (const float*)d_in[4];
  const float* w1b  = (const float*)d_in[5];
  const float* b1b  = (const float*)d_in[6];
  const float* w2a  = (const float*)d_in[7];
  const float* b2a  = (const float*)d_in[8];
  const float* w2b  = (const float*)d_in[9];
  const float* b2b  = (const float*)d_in[10];
  const float* wc   = (const float*)d_in[11];
  const float* bc   = (const float*)d_in[12];
  float* out = (float*)d_out;

  const int* src = eidx;
  const int* dst = eidx + GIN_N_EDGES;

  const size_t nodeBufElems = (size_t)GIN_N_NODES * GIN_HID;  // 6.4M floats
  float* agg  = (float*)d_ws;
  float* bufA = agg + nodeBufElems;
  float* bufB = bufA + nodeBufElems;
  float* g    = bufB + nodeBufElems;  // 64*128 floats

  const int mBlocks = GIN_N_NODES / GIN_ROWS_BLK;  // 625

  // ---- layer 1 ----
  gin_zero_f32<<<1024, 256, 0, stream>>>(agg, (int)(nodeBufElems / 4));
  gin_scatter_edges<<<25000, 256, 0, stream>>>(x, src, dst, agg);
  gin_gemm_relu<true><<<mBlocks, 256, 0, stream>>>(x, agg, w1a, b1a, bufA);
  gin_gemm_relu<false><<<mBlocks, 256, 0, stream>>>(bufA, nullptr, w1b, b1b, bufB);

  // ---- layer 2 ----
  gin_zero_f32<<<1024, 256, 0, stream>>>(agg, (int)(nodeBufElems / 4));
  gin_scatter_edges<<<25000, 256, 0, stream>>>(bufB, src, dst, agg);
  gin_gemm_relu<true><<<mBlocks, 256, 0, stream>>>(bufB, agg, w2a, b2a, bufA);
  gin_gemm_relu<false><<<mBlocks, 256, 0, stream>>>(bufA, nullptr, w2b, b2b, bufB);

  // ---- pool + classify ----
  gin_zero_f32<<<8, 256, 0, stream>>>(g, (GIN_N_GRAPHS * GIN_HID) / 4);
  gin_pool<<<6250, 256, 0, stream>>>(bufB, batch, g);
  gin_classifier<<<3, 256, 0, stream>>>(g, wc, bc, out);
}